// OneToOneLinear_7146825580919
// MI455X (gfx1250) — compile-verified
//
#include <hip/hip_runtime.h>
#include <hip/hip_bf16.h>

// out[n] = prod_j sigmoid(10*(x[n,j]*w[j]+b[j]))^(1/2)
//        = exp(-0.5 * sum_j softplus(-z_j)),  z_j = 10*(x*w+b)
//
// Computed entirely in the log2 domain (v_exp_f32/v_log_f32 are base-2 on AMD):
//   t_j   = (10*log2e)*(x*w+b)
//   u2_j  = softplus(-z_j)/ln2 = max(-t_j,0) + log2(1 + 2^(-|t_j|))
//   out   = 2^(-0.5 * sum_j u2_j)
//
// Row-reduction done with V_WMMA_F32_16X16X4_F32:
//   A = ones(16x4), B[k][n] = u2 values (4x16)  =>  D[m][n] = sum_k B[k][n]
// chained over 16 K-chunks => D[.][n] = full 64-feature sum for row n, broadcast
// across m, so lane n reads acc[0] directly (no shuffles, exact fp32 accumulation).

typedef float v2f __attribute__((ext_vector_type(2)));
typedef float v8f __attribute__((ext_vector_type(8)));

#define FEAT 64
#define ROWS_PER_WAVE 16
#define WAVES_PER_BLOCK 8
#define BLOCK_THREADS (WAVES_PER_BLOCK * 32)

#define ZSCALE 14.4269504088896340736f   /* 10 * log2(e) */

// softplus(-z)/ln2 with t = z*log2e, using raw base-2 hardware transcendentals:
// one v_exp_f32 + one v_log_f32, -|t| folds into a source modifier.
__device__ __forceinline__ float softplus_neg_log2(float t) {
    float e = __builtin_amdgcn_exp2f(-__builtin_fabsf(t)); // 2^(-|t|)  in (0,1]
    float l = __builtin_amdgcn_logf(1.0f + e);             // log2(1+e) in [0,1]
    return fmaxf(-t, 0.0f) + l;
}

__global__ __launch_bounds__(BLOCK_THREADS)
void onetoone_prod_kernel(const float* __restrict__ x,
                          const float* __restrict__ w,
                          const float* __restrict__ b,
                          float* __restrict__ out,
                          int nRows)
{
    const int lane = threadIdx.x & 31;
    const int wave = threadIdx.x >> 5;
    const int n    = lane & 15;   // tile row / B-matrix column
    const int half = lane >> 4;   // 0 -> K{0,1}, 1 -> K{2,3}

    const long tile    = (long)blockIdx.x * WAVES_PER_BLOCK + wave;
    const long rowBase = tile * ROWS_PER_WAVE;
    if (rowBase >= (long)nRows) return;   // whole-wave exit; EXEC all-ones below

    const float* rowPtr = x + (rowBase + n) * FEAT + 2 * half;
    const float* wPtr   = w + 2 * half;
    const float* bPtr   = b + 2 * half;

    v8f acc = {};               // C/D accumulator (f32, exact)
    v2f ones;
    ones.x = 1.0f; ones.y = 1.0f;   // A = ones(16x4): both A VGPRs are 1.0 in all lanes

    #pragma unroll
    for (int kk = 0; kk < 16; ++kk) {
        // lane n   : features 4kk+0, 4kk+1 of row (rowBase+n)   -> B[K=0/1][n]
        // lane n+16: features 4kk+2, 4kk+3 of row (rowBase+n)   -> B[K=2/3][n]
        v2f xv = *(const v2f*)(rowPtr + 4 * kk);
        v2f wv = *(const v2f*)(wPtr   + 4 * kk);
        v2f bv = *(const v2f*)(bPtr   + 4 * kk);

        float t0 = ZSCALE * fmaf(xv.x, wv.x, bv.x);
        float t1 = ZSCALE * fmaf(xv.y, wv.y, bv.y);

        v2f u;
        u.x = softplus_neg_log2(t0);
        u.y = softplus_neg_log2(t1);

        // D = A(ones) * B(u) + C : per-column K-sum, accumulated over 16 chunks
        acc = __builtin_amdgcn_wmma_f32_16x16x4_f32(
                  /*neg_a=*/false, ones,
                  /*neg_b=*/false, u,
                  /*c_mod=*/(short)0, acc,
                  /*reuse_a=*/false, /*reuse_b=*/false);
    }

    // Every m-row of D holds the same column sums; lane n, VGPR0 = sum for row n.
    float S = acc[0];
    if (half == 0) {
        out[rowBase + n] = __builtin_amdgcn_exp2f(-0.5f * S);
    }
}

extern "C" void kernel_launch(void* const* d_in, const int* in_sizes, int n_in,
                              void* d_out, int out_size, void* d_ws, size_t ws_size,
                              hipStream_t stream) {
    const float* x = (const float*)d_in[0];   // [N, 64] fp32
    const float* w = (const float*)d_in[1];   // [64]    fp32
    const float* b = (const float*)d_in[2];   // [64]    fp32
    float* out = (float*)d_out;               // [N]     fp32

    const int nRows = in_sizes[0] / FEAT;
    const long tiles  = ((long)nRows + ROWS_PER_WAVE - 1) / ROWS_PER_WAVE;
    const long blocks = (tiles + WAVES_PER_BLOCK - 1) / WAVES_PER_BLOCK;

    onetoone_prod_kernel<<<(int)blocks, BLOCK_THREADS, 0, stream>>>(x, w, b, out, nRows);
}